// EdgePoolBlock_10917806867170
// MI455X (gfx1250) — compile-verified
//
#include <hip/hip_runtime.h>
#include <hip/hip_bf16.h>

// ---------------------------------------------------------------------------
// EdgePoolBlock for MI455X (gfx1250, wave32, WMMA).
// All heavy matmuls use V_WMMA_F32_16X16X4_F32 (full fp32 precision on the
// matrix pipe).  Algebraic reductions turn the O(n^3 d) reference into a
// handful of small GEMMs + a 32-sparse attention; the kernel is dominated by
// writing the 75MB pair_attn output (L2/HBM-bound).
// ---------------------------------------------------------------------------

#define N_NODES 384
#define D_DIM   128
#define DEG     32
#define E_EDGES (N_NODES * DEG)      // 12288
#define K_SEL   (E_EDGES / 2)        // 6144
#define NPAD    16384                // bitonic pad
#define PH      2
#define PDH     64
#define GH      4

typedef float v2f __attribute__((ext_vector_type(2)));
typedef float v8f __attribute__((ext_vector_type(8)));

// ---------------------------------------------------------------------------
// Generic fp32 WMMA GEMM: C[M,N] = act(A[M,K] @ B(K,N) + bias), optional B^T.
// TRANSB / DORELU are template params so the inner K-loop is branch-free.
// Grid: (M/16, N/64), block 128 (4 waves, one 16x16 tile per wave).
// Fragment layout per CDNA5 ISA 7.12.2 (32-bit A 16x4): lanes 0-15 carry
// K={0,1} in the two VGPRs, lanes 16-31 carry K={2,3}.  B symmetric
// (lane%16 -> N, lane/16 selects K pair).  C: vgpr r -> row r (+8 for upper
// lane half), lane%16 -> col.
// ---------------------------------------------------------------------------
template <int TRANSB, int DORELU>
__global__ __launch_bounds__(128) void gemm_f32_wmma(
    const float* __restrict__ A, int lda,
    const float* __restrict__ B, int ldb,
    const float* __restrict__ bias,
    float* __restrict__ C, int ldc,
    int M, int N, int K) {
  const int wave = threadIdx.x >> 5;
  const int lane = threadIdx.x & 31;
  const int half = lane >> 4;
  const int l15  = lane & 15;
  const int m0 = blockIdx.x * 16;
  const int n0 = blockIdx.y * 64 + wave * 16;
  if (m0 >= M || n0 >= N) return;       // wave-uniform

  v8f acc = {};
  const float* Arow = A + (long)(m0 + l15) * lda + (half << 1);
  const float* Bptr;
  if (TRANSB) {
    Bptr = B + (long)(n0 + l15) * ldb + (half << 1);   // advance by k
  } else {
    Bptr = B + (long)(half << 1) * ldb + n0 + l15;     // advance by k*ldb
  }
#pragma unroll 4
  for (int k0 = 0; k0 < K; k0 += 4) {
    v2f a, b;
    a.x = Arow[k0];
    a.y = Arow[k0 + 1];
    if (TRANSB) {
      b.x = Bptr[k0];
      b.y = Bptr[k0 + 1];
    } else {
      b.x = Bptr[(long)k0 * ldb];
      b.y = Bptr[(long)(k0 + 1) * ldb];
    }
    acc = __builtin_amdgcn_wmma_f32_16x16x4_f32(false, a, false, b,
                                                (short)0, acc, false, false);
  }
  const int n = n0 + l15;
  const float bv = bias ? bias[n] : 0.0f;
#pragma unroll
  for (int r = 0; r < 8; ++r) {
    const int m = m0 + r + (half << 3);
    float v = acc[r] + bv;
    if (DORELU) v = fmaxf(v, 0.0f);
    C[(long)m * ldc + n] = v;
  }
}

// ---------------------------------------------------------------------------
// LayerNorm over rows of x (384 x 128).
// ---------------------------------------------------------------------------
__global__ __launch_bounds__(128) void ln_kernel(
    const float* __restrict__ x, const float* __restrict__ g,
    const float* __restrict__ b, float* __restrict__ out) {
  const int row = blockIdx.x, tid = threadIdx.x;
  __shared__ float red[128];
  __shared__ float mu_s, var_s;
  const float v = x[row * D_DIM + tid];
  red[tid] = v; __syncthreads();
  for (int s = 64; s > 0; s >>= 1) { if (tid < s) red[tid] += red[tid + s]; __syncthreads(); }
  if (tid == 0) mu_s = red[0] * (1.0f / D_DIM);
  __syncthreads();
  const float d = v - mu_s;
  red[tid] = d * d; __syncthreads();
  for (int s = 64; s > 0; s >>= 1) { if (tid < s) red[tid] += red[tid + s]; __syncthreads(); }
  if (tid == 0) var_s = red[0] * (1.0f / D_DIM);
  __syncthreads();
  out[row * D_DIM + tid] = d * rsqrtf(var_s + 1e-5f) * g[tid] + b[tid];
}

// ---------------------------------------------------------------------------
// Small per-row dot products: A_h[i] = qi_h[i].ki_h[i];  Wws_h[k] = W_h[k].ws
// ---------------------------------------------------------------------------
__global__ void small_dots_kernel(const float* __restrict__ qi,
                                  const float* __restrict__ ki,
                                  const float* __restrict__ W,
                                  const float* __restrict__ wsv,
                                  float* __restrict__ Ah,
                                  float* __restrict__ Wws) {
  const int u = blockIdx.x * blockDim.x + threadIdx.x;
  if (u >= 2 * N_NODES) return;
  const int h = u / N_NODES, i = u % N_NODES;
  float s = 0.f;
  for (int d = 0; d < PDH; ++d)
    s += qi[i * D_DIM + h * PDH + d] * ki[i * D_DIM + h * PDH + d];
  Ah[u] = s;
  float t = 0.f;
  for (int d = 0; d < D_DIM; ++d)
    t += W[(long)h * (N_NODES * D_DIM) + i * D_DIM + d] * wsv[d];
  Wws[u] = t;
}

// ---------------------------------------------------------------------------
// Per-i attention + pair_attn / pair_edge.  One block per i, 256 threads.
// attn is nonzero only at the 32 masked neighbors of i; softmax over those.
// Out(64x128 qj-tile) = viwo[i] + At(64x64) @ Wg(64x128) via fp32 WMMA.
// ---------------------------------------------------------------------------
__global__ __launch_bounds__(256) void attn_pair_kernel(
    const float* __restrict__ Bsc, const float* __restrict__ Csc,
    const float* __restrict__ Dsc, const float* __restrict__ Ah,
    const float* __restrict__ W,   const float* __restrict__ Wws,
    const float* __restrict__ viwo,
    const int* __restrict__ dstv,
    const float* __restrict__ wsv, const float* __restrict__ bsv,
    float* __restrict__ pair_attn_out, float* __restrict__ pair_edge_out) {
  const int i = blockIdx.x;
  const int tid = threadIdx.x;
  __shared__ int   nbr[DEG];
  __shared__ float Wg[64][D_DIM];     // rows 0..31 = W0[nbr], 32..63 = W1[nbr]
  __shared__ float WwsS[64];
  __shared__ float At[64][64];        // attn tile, K index = h*32+t
  __shared__ float viwoS[D_DIM];
  __shared__ float BsS[2][DEG];
  __shared__ float AhS[2];
  __shared__ float basePe;

  if (tid < DEG) nbr[tid] = dstv[i * DEG + tid];
  __syncthreads();
  for (int idx = tid; idx < 64 * D_DIM; idx += 256) {
    const int row = idx >> 7, col = idx & 127;
    const int h = row >> 5, t = row & 31;
    Wg[row][col] = W[(long)h * (N_NODES * D_DIM) + nbr[t] * D_DIM + col];
  }
  if (tid < 64) {
    const int h = tid >> 5, t = tid & 31;
    WwsS[tid] = Wws[h * N_NODES + nbr[t]];
    BsS[h][t] = Bsc[(long)h * (N_NODES * N_NODES) + i * N_NODES + nbr[t]];
  }
  if (tid < D_DIM) viwoS[tid] = viwo[i * D_DIM + tid];
  if (tid < 2) AhS[tid] = Ah[tid * N_NODES + i];
  __syncthreads();
  if (tid == 0) {
    float s = bsv[0];
    for (int d = 0; d < D_DIM; ++d) s += viwoS[d] * wsv[d];
    basePe = s;
  }
  __syncthreads();

  const int wave = tid >> 5, lane = tid & 31;
  const int half = lane >> 4, l15 = lane & 15;

  for (int q0 = 0; q0 < N_NODES; q0 += 64) {
    // (a) masked softmax over 32 neighbors for 64 qj x 2 heads
    if (tid < 128) {
      const int ql = tid >> 1;        // local qj
      const int h  = tid & 1;
      const int q  = q0 + ql;
      const float base = AhS[h] + Csc[(long)h * (N_NODES * N_NODES) + q * N_NODES + i];
      const float* Drow = Dsc + (long)h * (N_NODES * N_NODES) + (long)q * N_NODES;
      float sc[DEG];
      float mx = -3.0e38f;
#pragma unroll
      for (int t = 0; t < DEG; ++t) {
        const float s = (base + BsS[h][t] + Drow[nbr[t]]) * 0.125f; // PDH^-0.5
        sc[t] = s;
        mx = fmaxf(mx, s);
      }
      float sum = 0.f;
#pragma unroll
      for (int t = 0; t < DEG; ++t) { const float e = expf(sc[t] - mx); sc[t] = e; sum += e; }
      const float inv = 1.0f / sum;
#pragma unroll
      for (int t = 0; t < DEG; ++t) At[ql][h * DEG + t] = sc[t] * inv;
    }
    __syncthreads();

    // (c) WMMA: Out(64x128) = At(64x64) @ Wg(64x128); 32 tiles over 8 waves
    for (int tt = 0; tt < 4; ++tt) {
      const int tile = wave * 4 + tt;
      const int tr = tile >> 3;       // 0..3
      const int tc = tile & 7;        // 0..7
      v8f acc = {};
#pragma unroll
      for (int k0 = 0; k0 < 64; k0 += 4) {
        const int ka = k0 + (half << 1);
        v2f a, b;
        a.x = At[tr * 16 + l15][ka];
        a.y = At[tr * 16 + l15][ka + 1];
        b.x = Wg[ka][tc * 16 + l15];
        b.y = Wg[ka + 1][tc * 16 + l15];
        acc = __builtin_amdgcn_wmma_f32_16x16x4_f32(false, a, false, b,
                                                    (short)0, acc, false, false);
      }
      const int n = tc * 16 + l15;
      const float add = viwoS[n];
#pragma unroll
      for (int r = 0; r < 8; ++r) {
        const int qrow = q0 + tr * 16 + r + (half << 3);
        pair_attn_out[((long)i * N_NODES + qrow) * D_DIM + n] = acc[r] + add;
      }
    }

    // (b) pair_edge for this qj tile
    if (tid < 64) {
      float s = basePe;
#pragma unroll
      for (int kk = 0; kk < 64; ++kk) s += At[tid][kk] * WwsS[kk];
      pair_edge_out[(long)i * N_NODES + q0 + tid] = 1.0f / (1.0f + expf(-s));
    }
    __syncthreads();
  }
}

// ---------------------------------------------------------------------------
// Edge scores -> sort keys.  key = ord(score)<<32 | (~idx) so that a
// descending u64 sort reproduces JAX top_k order (value desc, index asc).
// ---------------------------------------------------------------------------
__device__ __forceinline__ unsigned ord_encode(float v) {
  const unsigned b = __float_as_uint(v);
  return (b & 0x80000000u) ? ~b : (b | 0x80000000u);
}
__device__ __forceinline__ float ord_decode(unsigned o) {
  const unsigned b = (o & 0x80000000u) ? (o & 0x7FFFFFFFu) : ~o;
  return __uint_as_float(b);
}

__global__ void score_key_kernel(const float* __restrict__ pair_edge,
                                 const int* __restrict__ src,
                                 const int* __restrict__ dstv,
                                 const float* __restrict__ sel_w,
                                 unsigned long long* __restrict__ keys) {
  const int e = blockIdx.x * blockDim.x + threadIdx.x;
  if (e >= NPAD) return;
  if (e < E_EDGES) {
    const float sw = sel_w[0];
    const float nrm = sqrtf(sw * sw);
    const float filt = pair_edge[(long)src[e] * N_NODES + dstv[e]];
    const float score = tanhf(filt * sw / nrm);
    keys[e] = ((unsigned long long)ord_encode(score) << 32) |
              (unsigned long long)(0xFFFFFFFFu - (unsigned)e);
  } else {
    keys[e] = 0ull;   // smaller than any real key -> sorts last (descending)
  }
}

__global__ void bitonic_step(unsigned long long* __restrict__ keys, int j, int k) {
  const int idx = blockIdx.x * blockDim.x + threadIdx.x;
  if (idx >= NPAD) return;
  const int l = idx ^ j;
  if (l > idx) {
    const unsigned long long a = keys[idx], b = keys[l];
    const bool up = ((idx & k) == 0);          // descending overall
    if (up ? (a < b) : (a > b)) { keys[idx] = b; keys[l] = a; }
  }
}

__global__ void select_kernel(const unsigned long long* __restrict__ keys,
                              const int* __restrict__ src,
                              const int* __restrict__ dstv,
                              float* __restrict__ out_ei,
                              float* __restrict__ out_vals,
                              float* __restrict__ out_perm,
                              float* __restrict__ out_vals2,
                              int* __restrict__ s2, int* __restrict__ d2,
                              float* __restrict__ valf) {
  const int e = blockIdx.x * blockDim.x + threadIdx.x;
  if (e >= K_SEL) return;
  const unsigned long long key = keys[e];
  const unsigned idx = 0xFFFFFFFFu - (unsigned)(key & 0xFFFFFFFFull);
  const float val = ord_decode((unsigned)(key >> 32));
  const int s = src[idx], d = dstv[idx];
  out_ei[e] = (float)s;
  out_ei[K_SEL + e] = (float)d;
  out_vals[e] = val;
  out_perm[e] = (float)idx;
  out_vals2[e] = val;
  s2[e] = s; d2[e] = d; valf[e] = val;
}

// ---------------------------------------------------------------------------
// GAT: alpha[e,g] = sum_d leaky_relu(xl[s2]+xr[d2]+val*lew, 0.2) * att[g]
// ---------------------------------------------------------------------------
__global__ void alpha_kernel(const float* __restrict__ xl,
                             const float* __restrict__ xr,
                             const float* __restrict__ valf,
                             const int* __restrict__ s2,
                             const int* __restrict__ d2,
                             const float* __restrict__ lew,
                             const float* __restrict__ att,
                             float* __restrict__ alpha) {
  const int u = blockIdx.x * blockDim.x + threadIdx.x;
  if (u >= K_SEL * GH) return;
  const int e = u >> 2, g = u & 3;
  const float* XL = xl + (long)s2[e] * (GH * D_DIM) + g * D_DIM;
  const float* XR = xr + (long)d2[e] * (GH * D_DIM) + g * D_DIM;
  const float* LE = lew + g * D_DIM;
  const float* AT = att + g * D_DIM;
  const float v = valf[e];
  float s = 0.f;
  for (int d = 0; d < D_DIM; ++d) {
    float m = XL[d] + XR[d] + v * LE[d];
    m = (m > 0.f) ? m : 0.2f * m;
    s += m * AT[d];
  }
  alpha[u] = s;
}

// ---------------------------------------------------------------------------
// Deterministic per-dst-node segment softmax + aggregation + residual.
// One block per node (<=32 incoming selected edges per node by construction).
// ---------------------------------------------------------------------------
__global__ __launch_bounds__(128) void gat_agg_kernel(
    const float* __restrict__ xl, const int* __restrict__ s2,
    const int* __restrict__ d2,  const float* __restrict__ alpha,
    const float* __restrict__ x, const float* __restrict__ gat_bias,
    float* __restrict__ x_out) {
  const int nid = blockIdx.x, tid = threadIdx.x;
  __shared__ int elist[DEG];
  __shared__ int ecnt;
  __shared__ float aW[DEG][GH];
  if (tid == 0) {
    int c = 0;
    for (int e = 0; e < K_SEL; ++e)
      if (d2[e] == nid && c < DEG) elist[c++] = e;
    ecnt = c;
  }
  __syncthreads();
  const int cnt = ecnt;
  if (tid < GH) {
    const int g = tid;
    float mx = -3.0e38f;
    for (int t = 0; t < cnt; ++t) mx = fmaxf(mx, alpha[elist[t] * GH + g]);
    if (cnt == 0) mx = 0.f;                       // isfinite() fixup
    float den = 0.f;
    for (int t = 0; t < cnt; ++t) {
      const float e = expf(alpha[elist[t] * GH + g] - mx);
      aW[t][g] = e; den += e;
    }
    const float inv = 1.0f / (den + 1e-16f);
    for (int t = 0; t < cnt; ++t) aW[t][g] *= inv;
  }
  __syncthreads();
  float acc = 0.f;
  for (int g = 0; g < GH; ++g) {
    float s = 0.f;
    for (int t = 0; t < cnt; ++t)
      s += xl[(long)s2[elist[t]] * (GH * D_DIM) + g * D_DIM + tid] * aW[t][g];
    acc += s;
  }
  const float out = acc * (1.0f / GH) + gat_bias[tid];
  x_out[(long)nid * D_DIM + tid] = x[(long)nid * D_DIM + tid] + out;
}

// ---------------------------------------------------------------------------
// Workspace layout (float element offsets)
// ---------------------------------------------------------------------------
#define OFF_LN    0L
#define OFF_HMID  49152L
#define OFF_H     245760L
#define OFF_QI    344064L
#define OFF_QO    393216L
#define OFF_KI    442368L
#define OFF_KO    491520L
#define OFF_VI    540672L
#define OFF_VO    589824L
#define OFF_VIWO  638976L
#define OFF_W     688128L
#define OFF_B     786432L
#define OFF_C     1081344L
#define OFF_D     1376256L
#define OFF_AH    1671168L
#define OFF_WWS   1671936L
#define OFF_KEYS  1672704L   // 16384 u64 (8B aligned)
#define OFF_S2    1705472L
#define OFF_D2    1711616L
#define OFF_VALF  1717760L
#define OFF_XL    1723904L
#define OFF_XR    1920512L
#define OFF_ALPHA 2117120L

extern "C" void kernel_launch(void* const* d_in, const int* in_sizes, int n_in,
                              void* d_out, int out_size, void* d_ws, size_t ws_size,
                              hipStream_t stream) {
  const float* x    = (const float*)d_in[0];
  const int*   ei   = (const int*)d_in[1];
  const float* ln_g = (const float*)d_in[2];
  const float* ln_b = (const float*)d_in[3];
  const float* w1   = (const float*)d_in[4];
  const float* b1   = (const float*)d_in[5];
  const float* w2   = (const float*)d_in[6];
  const float* b2   = (const float*)d_in[7];
  const float* wq   = (const float*)d_in[8];
  const float* wk   = (const float*)d_in[9];
  const float* wv   = (const float*)d_in[10];
  const float* wo   = (const float*)d_in[11];
  const float* wsv  = (const float*)d_in[12];
  const float* bsv  = (const float*)d_in[13];
  const float* selw = (const float*)d_in[14];
  const float* llw  = (const float*)d_in[15];
  const float* llb  = (const float*)d_in[16];
  const float* lrw  = (const float*)d_in[17];
  const float* lrb  = (const float*)d_in[18];
  const float* att  = (const float*)d_in[19];
  const float* lew  = (const float*)d_in[20];
  const float* gbias= (const float*)d_in[21];

  const int* src  = ei;              // first row
  const int* dstv = ei + E_EDGES;    // second row

  float* W  = (float*)d_ws;
  float* ln = W + OFF_LN;
  float* hmid = W + OFF_HMID;
  float* h  = W + OFF_H;
  float* qi = W + OFF_QI; float* qo = W + OFF_QO;
  float* ki = W + OFF_KI; float* ko = W + OFF_KO;
  float* vi = W + OFF_VI; float* vo = W + OFF_VO;
  float* viwo = W + OFF_VIWO;
  float* Wmat = W + OFF_W;
  float* Bsc  = W + OFF_B;
  float* Csc  = W + OFF_C;
  float* Dsc  = W + OFF_D;
  float* Ah   = W + OFF_AH;
  float* Wws  = W + OFF_WWS;
  unsigned long long* keys = (unsigned long long*)(W + OFF_KEYS);
  int* s2 = (int*)(W + OFF_S2);
  int* d2 = (int*)(W + OFF_D2);
  float* valf = W + OFF_VALF;
  float* xl = W + OFF_XL;
  float* xr = W + OFF_XR;
  float* alpha = W + OFF_ALPHA;

  float* out = (float*)d_out;
  float* out_xout  = out + 0;
  float* out_ei    = out + 49152;
  float* out_vals  = out + 61440;
  float* out_perm  = out + 67584;
  float* out_vals2 = out + 73728;
  float* out_pedge = out + 79872;
  float* out_pattn = out + 227328;

  const dim3 blk(128);
  // 1. LayerNorm
  ln_kernel<<<N_NODES, 128, 0, stream>>>(x, ln_g, ln_b, ln);
  // 2. MLP
  gemm_f32_wmma<0, 1><<<dim3(24, 8), blk, 0, stream>>>(ln, 128, w1, 512, b1, hmid, 512, 384, 512, 128);
  gemm_f32_wmma<0, 0><<<dim3(24, 4), blk, 0, stream>>>(hmid, 512, w2, 256, b2, h, 256, 384, 256, 512);
  // 3. q/k/v projections of ingoing (h[:, :128]) and outgoing (h[:, 128:])
  gemm_f32_wmma<0, 0><<<dim3(24, 2), blk, 0, stream>>>(h,       256, wq, 128, nullptr, qi, 128, 384, 128, 128);
  gemm_f32_wmma<0, 0><<<dim3(24, 2), blk, 0, stream>>>(h + 128, 256, wq, 128, nullptr, qo, 128, 384, 128, 128);
  gemm_f32_wmma<0, 0><<<dim3(24, 2), blk, 0, stream>>>(h,       256, wk, 128, nullptr, ki, 128, 384, 128, 128);
  gemm_f32_wmma<0, 0><<<dim3(24, 2), blk, 0, stream>>>(h + 128, 256, wk, 128, nullptr, ko, 128, 384, 128, 128);
  gemm_f32_wmma<0, 0><<<dim3(24, 2), blk, 0, stream>>>(h,       256, wv, 128, nullptr, vi, 128, 384, 128, 128);
  gemm_f32_wmma<0, 0><<<dim3(24, 2), blk, 0, stream>>>(h + 128, 256, wv, 128, nullptr, vo, 128, 384, 128, 128);
  // 4. viwo = vi @ wo ;  W_h = vo[:,h*64:] @ wo[h*64:,:]
  gemm_f32_wmma<0, 0><<<dim3(24, 2), blk, 0, stream>>>(vi, 128, wo, 128, nullptr, viwo, 128, 384, 128, 128);
  gemm_f32_wmma<0, 0><<<dim3(24, 2), blk, 0, stream>>>(vo,      128, wo,            128, nullptr, Wmat,         128, 384, 128, 64);
  gemm_f32_wmma<0, 0><<<dim3(24, 2), blk, 0, stream>>>(vo + 64, 128, wo + 64 * 128, 128, nullptr, Wmat + 49152, 128, 384, 128, 64);
  // 5. score component matrices per head: B = qi_h @ ko_h^T, C = qo_h @ ki_h^T, D = qo_h @ ko_h^T
  for (int hh = 0; hh < 2; ++hh) {
    const long o = (long)hh * (N_NODES * N_NODES);
    gemm_f32_wmma<1, 0><<<dim3(24, 6), blk, 0, stream>>>(qi + hh * 64, 128, ko + hh * 64, 128, nullptr, Bsc + o, 384, 384, 384, 64);
    gemm_f32_wmma<1, 0><<<dim3(24, 6), blk, 0, stream>>>(qo + hh * 64, 128, ki + hh * 64, 128, nullptr, Csc + o, 384, 384, 384, 64);
    gemm_f32_wmma<1, 0><<<dim3(24, 6), blk, 0, stream>>>(qo + hh * 64, 128, ko + hh * 64, 128, nullptr, Dsc + o, 384, 384, 384, 64);
  }
  // 6. diag dots + W.ws
  small_dots_kernel<<<3, 256, 0, stream>>>(qi, ki, Wmat, wsv, Ah, Wws);
  // 7. attention -> pair_attn / pair_edge
  attn_pair_kernel<<<N_NODES, 256, 0, stream>>>(Bsc, Csc, Dsc, Ah, Wmat, Wws, viwo,
                                                dstv, wsv, bsv, out_pattn, out_pedge);
  // 8. edge scores -> keys; exact top-k via bitonic sort (deterministic)
  score_key_kernel<<<NPAD / 256, 256, 0, stream>>>(out_pedge, src, dstv, selw, keys);
  for (int k = 2; k <= NPAD; k <<= 1)
    for (int j = k >> 1; j > 0; j >>= 1)
      bitonic_step<<<NPAD / 256, 256, 0, stream>>>(keys, j, k);
  select_kernel<<<K_SEL / 256, 256, 0, stream>>>(keys, src, dstv, out_ei, out_vals,
                                                 out_perm, out_vals2, s2, d2, valf);
  // 9. GAT: xl / xr projections
  gemm_f32_wmma<0, 0><<<dim3(24, 8), blk, 0, stream>>>(x, 128, llw, 512, llb, xl, 512, 384, 512, 128);
  gemm_f32_wmma<0, 0><<<dim3(24, 8), blk, 0, stream>>>(x, 128, lrw, 512, lrb, xr, 512, 384, 512, 128);
  // 10. edge logits, segment softmax + aggregation + residual
  alpha_kernel<<<(K_SEL * GH) / 256, 256, 0, stream>>>(xl, xr, valf, s2, d2, lew, att, alpha);
  gat_agg_kernel<<<N_NODES, 128, 0, stream>>>(xl, s2, d2, alpha, x, gbias, out_xout);

  (void)in_sizes; (void)n_in; (void)out_size; (void)ws_size;
}